// ConfigurableGCN_75746043232585
// MI455X (gfx1250) — compile-verified
//
#include <hip/hip_runtime.h>

#define FD 64  // feature dim

typedef __attribute__((ext_vector_type(2))) float v2f;
typedef __attribute__((ext_vector_type(8))) float v8f;

// ---------------------------------------------------------------- degree prep
__global__ void k_deg_init(float* __restrict__ deg, int n) {
    int i = blockIdx.x * blockDim.x + threadIdx.x;
    if (i < n) deg[i] = 1.0f;  // self-loop contributes 1
}

__global__ void k_deg_accum(const int* __restrict__ dst, float* __restrict__ deg, int e) {
    int i = blockIdx.x * blockDim.x + threadIdx.x;
    if (i < e) atomicAdd(&deg[dst[i]], 1.0f);
}

__global__ void k_rsqrt_inplace(float* __restrict__ deg, int n) {
    int i = blockIdx.x * blockDim.x + threadIdx.x;
    if (i < n) deg[i] = rsqrtf(deg[i]);  // deg >= 1 guaranteed
}

// ---------------------------------------------------------------- buffer init
__global__ void k_zero(float* __restrict__ p, int count) {
    int i = blockIdx.x * blockDim.x + threadIdx.x;
    if (i < count) p[i] = 0.0f;
}

__global__ void k_bias_bcast(float* __restrict__ out, const float* __restrict__ b, int count) {
    int i = blockIdx.x * blockDim.x + threadIdx.x;
    if (i < count) out[i] = b[i & (FD - 1)];
}

// ---------------------------------------------------------------- WMMA GEMM
// O[N,64] = H[N,64] @ W[64,64], fp32 via v_wmma_f32_16x16x4_f32.
// One wave per 16x16 output tile; K-loop = 64/4 = 16 WMMA ops.
// A layout (16x4 f32, ISA 7.12.2): lanes 0-15 hold M=lane, VGPR{0,1}=K{0,1};
//                                  lanes 16-31 hold M=lane-16, VGPR{0,1}=K{2,3}.
// B layout (4x16 f32): lanes 0-15 hold N=lane, VGPR{0,1}=K{0,1}; lanes 16-31 K{2,3}.
// C/D layout (16x16 f32): VGPR v: lanes 0-15 -> M=v, lanes 16-31 -> M=8+v; N=lane&15.
__global__ __launch_bounds__(256)
void k_gemm_wmma(const float* __restrict__ H, const float* __restrict__ W,
                 float* __restrict__ O, int n) {
    const int wave = blockIdx.x * (blockDim.x >> 5) + (threadIdx.x >> 5);
    const int lane = threadIdx.x & 31;
    const int ntiles_m = (n + 15) >> 4;
    const int tm = wave >> 2;        // row tile   (FD/16 == 4 col tiles)
    const int tn = wave & 3;         // col tile
    if (tm >= ntiles_m) return;

    const int r0 = tm << 4;
    const int c0 = tn << 4;
    const int lrow = lane & 15;      // M (for A) / N (for B)
    const int hi   = lane >> 4;      // selects K pair {0,1} vs {2,3}

    int arow = r0 + lrow;
    if (arow >= n) arow = n - 1;     // clamp loads; stores guarded below

    v8f acc = {};
    #pragma unroll
    for (int k0 = 0; k0 < FD; k0 += 4) {
        const int k = k0 + 2 * hi;
        v2f a, b;
        const float* hp = H + (size_t)arow * FD + k;
        a.x = hp[0];
        a.y = hp[1];
        const float* wp = W + (size_t)k * FD + c0 + lrow;
        b.x = wp[0];
        b.y = wp[FD];
        acc = __builtin_amdgcn_wmma_f32_16x16x4_f32(
            /*neg_a=*/false, a, /*neg_b=*/false, b,
            /*c_mod=*/(short)0, acc, /*reuse_a=*/false, /*reuse_b=*/false);
    }

    #pragma unroll
    for (int v = 0; v < 8; ++v) {
        const int row = r0 + v + 8 * hi;
        if (row < n) O[(size_t)row * FD + c0 + lrow] = acc[v];
    }
}

// ---------------------------------------------------------------- edge scatter
// 16 threads per edge, float4 per thread: msg = ht[src]*dinv[src]*dinv[dst],
// atomically accumulated into agg[dst].
__global__ __launch_bounds__(256)
void k_scatter_edges(const int* __restrict__ src, const int* __restrict__ dst,
                     const float* __restrict__ dinv, const float* __restrict__ ht,
                     float* __restrict__ agg, int e) {
    const unsigned idx = blockIdx.x * blockDim.x + threadIdx.x;
    const unsigned eid = idx >> 4;
    if (eid >= (unsigned)e) return;
    const int part = idx & 15;

    const int s = src[eid];
    const int d = dst[eid];
    const float nrm = dinv[s] * dinv[d];

    const float4 v = *reinterpret_cast<const float4*>(ht + (size_t)s * FD + part * 4);
    float* out = agg + (size_t)d * FD + part * 4;
    atomicAdd(out + 0, v.x * nrm);
    atomicAdd(out + 1, v.y * nrm);
    atomicAdd(out + 2, v.z * nrm);
    atomicAdd(out + 3, v.w * nrm);
}

// Self-loop contribution: agg[i] += ht[i] * dinv[i]^2  (stream-ordered after
// the edge scatter, so plain RMW is race-free).
__global__ __launch_bounds__(256)
void k_self_loop(const float* __restrict__ dinv, const float* __restrict__ ht,
                 float* __restrict__ agg, int n) {
    const unsigned idx = blockIdx.x * blockDim.x + threadIdx.x;
    const unsigned i = idx >> 4;
    if (i >= (unsigned)n) return;
    const int part = idx & 15;
    const float w = dinv[i] * dinv[i];

    const float4 v = *reinterpret_cast<const float4*>(ht + (size_t)i * FD + part * 4);
    float4* ap = reinterpret_cast<float4*>(agg + (size_t)i * FD + part * 4);
    float4 a = *ap;
    a.x += v.x * w; a.y += v.y * w; a.z += v.z * w; a.w += v.w * w;
    *ap = a;
}

// ---------------------------------------------------------------- LN + ReLU
// One wave per row (D=64 -> 2 floats/lane), wave32 shuffle reduction.
__global__ __launch_bounds__(256)
void k_bias_ln_relu(const float* __restrict__ agg, const float* __restrict__ bias,
                    const float* __restrict__ g, const float* __restrict__ be,
                    float* __restrict__ out, int n) {
    const int row  = blockIdx.x * (blockDim.x >> 5) + (threadIdx.x >> 5);
    const int lane = threadIdx.x & 31;
    if (row >= n) return;

    const float* rp = agg + (size_t)row * FD;
    const float x0 = rp[lane]      + bias[lane];
    const float x1 = rp[lane + 32] + bias[lane + 32];

    float s  = x0 + x1;
    float s2 = x0 * x0 + x1 * x1;
    #pragma unroll
    for (int m = 16; m > 0; m >>= 1) {
        s  += __shfl_xor(s,  m, 32);
        s2 += __shfl_xor(s2, m, 32);
    }
    const float mu  = s * (1.0f / FD);
    const float var = s2 * (1.0f / FD) - mu * mu;
    const float rs  = rsqrtf(var + 1e-5f);

    const float y0 = (x0 - mu) * rs * g[lane]      + be[lane];
    const float y1 = (x1 - mu) * rs * g[lane + 32] + be[lane + 32];
    float* op = out + (size_t)row * FD;
    op[lane]      = fmaxf(y0, 0.0f);
    op[lane + 32] = fmaxf(y1, 0.0f);
}

// ---------------------------------------------------------------- launcher
extern "C" void kernel_launch(void* const* d_in, const int* in_sizes, int n_in,
                              void* d_out, int out_size, void* d_ws, size_t ws_size,
                              hipStream_t stream) {
    const float* x  = (const float*)d_in[0];
    const int*   ei = (const int*)d_in[1];
    const float* Wl[3]  = { (const float*)d_in[2], (const float*)d_in[4], (const float*)d_in[6] };
    const float* bl[3]  = { (const float*)d_in[3], (const float*)d_in[5], (const float*)d_in[7] };
    const float* gl[2]  = { (const float*)d_in[8], (const float*)d_in[10] };
    const float* bel[2] = { (const float*)d_in[9], (const float*)d_in[11] };

    const int n = in_sizes[0] / FD;   // 50000
    const int e = in_sizes[1] / 2;    // 800000
    const int* src = ei;
    const int* dst = ei + e;

    float* ws   = (float*)d_ws;
    float* dinv = ws;                               // n floats
    float* tmpG = dinv + n;                         // n*64  (GEMM output ht)
    float* tmpA = tmpG + (size_t)n * FD;            // n*64  (aggregation)
    float* tmpH = tmpA + (size_t)n * FD;            // n*64  (layer activations)

    const int B = 256;
    const int feat = n * FD;

    // Degrees -> dinv
    k_deg_init<<<(n + B - 1) / B, B, 0, stream>>>(dinv, n);
    k_deg_accum<<<(e + B - 1) / B, B, 0, stream>>>(dst, dinv, e);
    k_rsqrt_inplace<<<(n + B - 1) / B, B, 0, stream>>>(dinv, n);

    const int gemm_tiles  = ((n + 15) / 16) * (FD / 16);
    const int gemm_blocks = (gemm_tiles + 7) / 8;          // 8 waves per block
    const unsigned scat_threads = (unsigned)e * 16u;
    const unsigned self_threads = (unsigned)n * 16u;

    const float* hin = x;
    for (int l = 0; l < 3; ++l) {
        // Dense transform via fp32 WMMA
        k_gemm_wmma<<<gemm_blocks, B, 0, stream>>>(hin, Wl[l], tmpG, n);

        float* aggp;
        if (l < 2) {
            aggp = tmpA;
            k_zero<<<(feat + B - 1) / B, B, 0, stream>>>(aggp, feat);
        } else {
            aggp = (float*)d_out;                           // final: out = agg + b2
            k_bias_bcast<<<(feat + B - 1) / B, B, 0, stream>>>(aggp, bl[2], feat);
        }

        // Message passing (edges, then self-loops)
        k_scatter_edges<<<(scat_threads + B - 1) / B, B, 0, stream>>>(src, dst, dinv, tmpG, aggp, e);
        k_self_loop<<<(self_threads + B - 1) / B, B, 0, stream>>>(dinv, tmpG, aggp, n);

        if (l < 2) {
            k_bias_ln_relu<<<((n + 7) / 8), B, 0, stream>>>(aggp, bl[l], gl[l], bel[l], tmpH, n);
            hin = tmpH;
        }
    }
}